// L2Nonlinearity_50173807952920
// MI455X (gfx1250) — compile-verified
//
#include <hip/hip_runtime.h>

typedef float v2f __attribute__((ext_vector_type(2)));
typedef float v8f __attribute__((ext_vector_type(8)));

#define C2 1024  // 32*32

// Packed index for the WMMA A-operand layout of a [16 x 1024] fp32 matrix:
// tile kt = col/4 holds 64 floats; lane j<16 -> row j, K = l0,l0+1 ; lane j>=16 -> row j-16, K = l0+2,l0+3
__device__ __forceinline__ int pkIdx(int a, int col) {
  return (col >> 2) * 64 + ((col >> 1) & 1) * 32 + 2 * a + (col & 1);
}

// Repack a 1024x1024 row-major weight W (used as B = W^T, K=l, N=k) into
// per-tile WMMA B-operand order: [nt (64)][kt (256)][lane (32)] x float2.
__global__ __launch_bounds__(256) void repack_w(const float* __restrict__ W,
                                                float* __restrict__ Wp) {
  int t = blockIdx.x * 256 + threadIdx.x;  // one float2 per thread
  int lane = t & 31;
  int tile = t >> 5;          // nt*256 + kt
  int kt = tile & 255;
  int nt = tile >> 8;
  int row = nt * 16 + (lane & 15);        // N (output neuron k)
  int col = kt * 4 + (lane >> 4) * 2;     // K (input l)
  v2f v;
  v.x = W[row * C2 + col];
  v.y = W[row * C2 + col + 1];
  ((v2f*)Wp)[t] = v;
}

template <int D>
__global__ __launch_bounds__(256) void atom_block(
    const float* __restrict__ x, const int* __restrict__ idx,
    const float* __restrict__ S, const float* __restrict__ W1p,
    const float* __restrict__ b1, const float* __restrict__ g,
    const float* __restrict__ beta, const float* __restrict__ W2p,
    const float* __restrict__ b2, float* __restrict__ out) {
  __shared__ float yS[16 * D * 32];   // y tile, row-major [a][i][c]
  __shared__ float bufA[16 * C2];     // WMMA A-packed operand
  __shared__ float bufB[16 * C2];     // row-major scratch (Sy / h / l2')
  __shared__ float Ssh[D * D];
  __shared__ int   rowS[16 * D];
  __shared__ float muS[16], rsS[16];

  const int tid  = threadIdx.x;
  const int lane = tid & 31;
  const int wave = tid >> 5;
  const int a0   = blockIdx.x * 16;

  for (int k = tid; k < D * D; k += 256) Ssh[k] = S[k];
  for (int k = tid; k < 16 * D; k += 256) rowS[k] = idx[a0 * D + k];
  __syncthreads();
  for (int e = tid; e < 16 * D * 32; e += 256) {
    int m = e & 31, ai = e >> 5;
    yS[e] = x[(size_t)rowS[ai] * 32 + m];
  }
  __syncthreads();

  // Sy[a,i,m] = sum_j S[i,j] * y[a,j,m]  -> bufB
  for (int e = tid; e < 16 * D * 32; e += 256) {
    int m = e & 31, ai = e >> 5, i = ai % D, a = ai / D;
    float acc = 0.f;
    #pragma unroll
    for (int j = 0; j < D; ++j) acc += Ssh[i * D + j] * yS[(a * D + j) * 32 + m];
    bufB[e] = acc;
  }
  __syncthreads();

  // l2[a,c,m] = sum_i y[a,i,c] * Sy[a,i,m] -> bufA (A-packed)
  for (int e = tid; e < 16 * C2; e += 256) {
    int a = e >> 10, cm = e & 1023, c = cm >> 5, m = cm & 31;
    float acc = 0.f;
    #pragma unroll
    for (int i = 0; i < D; ++i)
      acc += yS[(a * D + i) * 32 + c] * bufB[(a * D + i) * 32 + m];
    bufA[pkIdx(a, cm)] = acc;
  }
  __syncthreads();

  const v2f* aPk = (const v2f*)bufA;

  // ---- GEMM1: h = l2 @ W1^T, wave computes 16 x 128 strip (8 tiles) ----
  {
    const v2f* bPk = (const v2f*)W1p;
    v8f acc[8] = {};
    for (int kt = 0; kt < 256; ++kt) {
      v2f av = aPk[kt * 32 + lane];
      #pragma unroll
      for (int n8 = 0; n8 < 8; ++n8) {
        int nt = wave * 8 + n8;
        v2f bv = bPk[(nt * 256 + kt) * 32 + lane];
        acc[n8] = __builtin_amdgcn_wmma_f32_16x16x4_f32(
            false, av, false, bv, (short)0, acc[n8], false, false);
      }
    }
    int ncol = lane & 15, nh = lane >> 4;
    #pragma unroll
    for (int n8 = 0; n8 < 8; ++n8) {
      int col = wave * 128 + n8 * 16 + ncol;
      float bias = b1[col];
      #pragma unroll
      for (int v = 0; v < 8; ++v) {
        int row = v + nh * 8;                 // D-matrix M index
        float p = acc[n8][v] + bias;
        bufB[row * C2 + col] = p / (1.f + __expf(-p));  // SiLU
      }
    }
  }
  __syncthreads();

  // ---- LayerNorm stats (2 rows per wave) ----
  for (int r = wave * 2; r < wave * 2 + 2; ++r) {
    float s = 0.f, sq = 0.f;
    for (int c = lane; c < C2; c += 32) {
      float v = bufB[r * C2 + c];
      s += v; sq += v * v;
    }
    #pragma unroll
    for (int off = 16; off > 0; off >>= 1) {
      s  += __shfl_down(s,  off, 32);
      sq += __shfl_down(sq, off, 32);
    }
    if (lane == 0) {
      float mu  = s * (1.f / 1024.f);
      float var = sq * (1.f / 1024.f) - mu * mu;
      muS[r] = mu;
      rsS[r] = rsqrtf(var + 1e-5f);
    }
  }
  __syncthreads();

  // normalize + scale/shift, rewrite into A-packed layout
  for (int e = tid; e < 16 * C2; e += 256) {
    int a = e >> 10, col = e & 1023;
    float v = (bufB[e] - muS[a]) * rsS[a] * g[col] + beta[col];
    bufA[pkIdx(a, col)] = v;
  }
  __syncthreads();

  // ---- GEMM2: l2' = h @ W2^T + b2 ----
  {
    const v2f* bPk = (const v2f*)W2p;
    v8f acc[8] = {};
    for (int kt = 0; kt < 256; ++kt) {
      v2f av = aPk[kt * 32 + lane];
      #pragma unroll
      for (int n8 = 0; n8 < 8; ++n8) {
        int nt = wave * 8 + n8;
        v2f bv = bPk[(nt * 256 + kt) * 32 + lane];
        acc[n8] = __builtin_amdgcn_wmma_f32_16x16x4_f32(
            false, av, false, bv, (short)0, acc[n8], false, false);
      }
    }
    int ncol = lane & 15, nh = lane >> 4;
    #pragma unroll
    for (int n8 = 0; n8 < 8; ++n8) {
      int col = wave * 128 + n8 * 16 + ncol;
      float bias = b2[col];
      #pragma unroll
      for (int v = 0; v < 8; ++v) {
        int row = v + nh * 8;
        bufB[row * C2 + col] = acc[n8][v] + bias;
      }
    }
  }
  __syncthreads();

  // ---- out[a,i,m] = sum_c y[a,i,c] * l2'[a,c,m], scatter ----
  for (int e = tid; e < 16 * D * 32; e += 256) {
    int m = e & 31, ai = e >> 5, a = ai / D;
    const float* yr = &yS[ai * 32];
    const float* lr = &bufB[a * C2 + m];
    float acc = 0.f;
    #pragma unroll
    for (int c = 0; c < 32; ++c) acc += yr[c] * lr[c * 32];
    out[(size_t)rowS[ai] * 32 + m] = acc;
  }
}

extern "C" void kernel_launch(void* const* d_in, const int* in_sizes, int n_in,
                              void* d_out, int out_size, void* d_ws, size_t ws_size,
                              hipStream_t stream) {
  const float* x  = (const float*)d_in[0];
  float* out = (float*)d_out;
  float* ws  = (float*)d_ws;

  const int counts[3] = {8192, 8192, 4096};
  float* wp[6];
  for (int i = 0; i < 6; ++i) wp[i] = ws + (size_t)i * C2 * C2;  // 24 MB total

  // One-time weight repack into WMMA B-operand layout (stream-ordered).
  for (int t = 0; t < 3; ++t) {
    const float* W1 = (const float*)d_in[1 + t * 8 + 2];
    const float* W2 = (const float*)d_in[1 + t * 8 + 6];
    repack_w<<<2048, 256, 0, stream>>>(W1, wp[2 * t]);
    repack_w<<<2048, 256, 0, stream>>>(W2, wp[2 * t + 1]);
  }

  for (int t = 0; t < 3; ++t) {
    const int*   idx = (const int*)d_in[1 + t * 8 + 0];
    const float* S   = (const float*)d_in[1 + t * 8 + 1];
    const float* b1  = (const float*)d_in[1 + t * 8 + 3];
    const float* g   = (const float*)d_in[1 + t * 8 + 4];
    const float* be  = (const float*)d_in[1 + t * 8 + 5];
    const float* b2  = (const float*)d_in[1 + t * 8 + 7];
    int blocks = counts[t] / 16;
    if (t == 0)
      atom_block<8><<<blocks, 256, 0, stream>>>(x, idx, S, wp[0], b1, g, be, wp[1], b2, out);
    else if (t == 1)
      atom_block<14><<<blocks, 256, 0, stream>>>(x, idx, S, wp[2], b1, g, be, wp[3], b2, out);
    else
      atom_block<18><<<blocks, 256, 0, stream>>>(x, idx, S, wp[4], b1, g, be, wp[5], b2, out);
  }
}